// ConformerBlock_89678917141134
// MI455X (gfx1250) — compile-verified
//
#include <hip/hip_runtime.h>
#include <hip/hip_bf16.h>
#include <math.h>

// ---------------------------------------------------------------------------
// Conformer block for MI455X (gfx1250, wave32, WMMA).
// All GEMMs go through V_WMMA_F32_16X16X4_F32 (fp32 in/out, matrix pipe).
// Banded longformer attention is also WMMA-based (scores and context).
// ---------------------------------------------------------------------------

typedef float v2f __attribute__((ext_vector_type(2)));
typedef float v8f __attribute__((ext_vector_type(8)));

static constexpr int Bb = 4;
static constexpr int Tt = 1024;
static constexpr int Dd = 512;
static constexpr int Hh = 8;
static constexpr int DHh = 64;
static constexpr int W1c = 32;
static constexpr int WBc = 65;
static constexpr int FFc = 2048;
static constexpr int CEc = 1024;
static constexpr int KSc = 31;

__device__ static inline v8f wmma4(v2f a, v2f b, v8f c) {
  // D = A(16x4, f32) * B(4x16, f32) + C(16x16, f32)
  return __builtin_amdgcn_wmma_f32_16x16x4_f32(
      /*neg_a=*/false, a, /*neg_b=*/false, b,
      /*c_mod=*/(short)0, c, /*reuse_a=*/false, /*reuse_b=*/false);
}

__device__ static inline float swish_f(float v) {
  return v / (1.0f + expf(-v));
}

// ---------------------------------------------------------------------------
// LayerNorm over D=512. One wave32 per row; 4 float4 per lane.
// ---------------------------------------------------------------------------
__global__ __launch_bounds__(256) void ln_kernel(
    const float* __restrict__ in, const float* __restrict__ g,
    const float* __restrict__ b, float* __restrict__ out, int rows) {
  int wave = threadIdx.x >> 5;
  int lane = threadIdx.x & 31;
  int row = blockIdx.x * (blockDim.x >> 5) + wave;
  if (row >= rows) return;
  const float4* x4 = (const float4*)(in + (size_t)row * Dd);
  float4 vals[4];
  float s = 0.f, s2 = 0.f;
#pragma unroll
  for (int i = 0; i < 4; ++i) {
    float4 t = x4[lane + i * 32];
    vals[i] = t;
    s += t.x + t.y + t.z + t.w;
    s2 += t.x * t.x + t.y * t.y + t.z * t.z + t.w * t.w;
  }
#pragma unroll
  for (int off = 16; off > 0; off >>= 1) {
    s += __shfl_xor(s, off, 32);
    s2 += __shfl_xor(s2, off, 32);
  }
  float m = s * (1.0f / Dd);
  float var = s2 * (1.0f / Dd) - m * m;
  float r = rsqrtf(var + 1e-5f);
  const float4* g4 = (const float4*)g;
  const float4* b4 = (const float4*)b;
  float4* o4 = (float4*)(out + (size_t)row * Dd);
#pragma unroll
  for (int i = 0; i < 4; ++i) {
    int c = lane + i * 32;
    float4 gg = g4[c], bb = b4[c], t = vals[i], o;
    o.x = (t.x - m) * r * gg.x + bb.x;
    o.y = (t.y - m) * r * gg.y + bb.y;
    o.z = (t.z - m) * r * gg.z + bb.z;
    o.w = (t.w - m) * r * gg.w + bb.w;
    o4[c] = o;
  }
}

// ---------------------------------------------------------------------------
// Tiled WMMA GEMM: out[M,N] = epilogue(A[M,K] @ W[K,N] + bias).
//   act: 0 = none, 1 = swish.  val = act(preScale*(acc+bias));
//   if res: val = res + resScale*val.
// 128x64 tile per workgroup, 8 waves; each wave owns a 16x64 strip
// (four 16x16 WMMA accumulators -> each A fragment feeds 4 WMMAs).
// A staged row-major stride 36 (16B-aligned rows, conflict-free b64 reads).
// B staged in (k-pair, n, 2) layout, k-pair stride 160 floats, so each
// B fragment is ONE aligned ds_load_b64 and the two lane-halves hit
// disjoint bank sets. Fragment loads are software-pipelined one k-step ahead.
// ---------------------------------------------------------------------------
__global__ __launch_bounds__(256) void gemm_wmma_kernel(
    const float* __restrict__ A, const float* __restrict__ W,
    const float* __restrict__ bias, float* __restrict__ out, int M, int N,
    int K, int act, float preScale, const float* __restrict__ res,
    float resScale) {
  constexpr int TM = 128, TN = 64, KC = 32;
  constexpr int ALD = 36;    // floats; 144B rows: 16B aligned, banks distinct
  constexpr int BPLD = 160;  // floats per k-pair row; +160 ≡ +32 banks
  __shared__ float As[TM * ALD];          // 18432 B
  __shared__ float Bs[(KC / 2) * BPLD];   // 10240 B

  const int tileN = blockIdx.x * TN;
  const int tileM = blockIdx.y * TM;
  const int tid = threadIdx.x;
  const int wave = tid >> 5;
  const int lane = tid & 31;
  const int lane_lo = lane & 15;
  const int half = lane >> 4;
  const int rw = wave * 16;  // wave's row offset in tile

  const float* Ablk = A + (size_t)tileM * K;
  const int aBase = (rw + lane_lo) * ALD + half * 2;  // + ks*4
  const int bBase = half * BPLD + lane_lo * 2;        // + ks*2*BPLD + nt*32

  v8f acc[4] = {};

  for (int k0 = 0; k0 < K; k0 += KC) {
    // ---- stage A tile: 128x32 floats as float4 (16B-aligned LDS stores) ----
#pragma unroll
    for (int i = 0; i < 4; ++i) {
      int lin = tid + i * 256;          // 1024 float4s
      int r = lin >> 3, c4 = lin & 7;
      float4 av = *(const float4*)(Ablk + (size_t)r * K + k0 + c4 * 4);
      *(float4*)(&As[r * ALD + c4 * 4]) = av;
    }
    // ---- stage B tile: 32x64 floats into (k-pair, n, 2) layout ----
#pragma unroll
    for (int i = 0; i < 2; ++i) {
      int lin = tid + i * 256;          // 512 float4s
      int r = lin >> 4, c4 = lin & 15;
      float4 bv = *(const float4*)(W + (size_t)(k0 + r) * N + tileN + c4 * 4);
      float* dst = &Bs[(r >> 1) * BPLD + (r & 1)];
      int n = c4 * 4;
      dst[(n + 0) * 2] = bv.x;
      dst[(n + 1) * 2] = bv.y;
      dst[(n + 2) * 2] = bv.z;
      dst[(n + 3) * 2] = bv.w;
    }
    // ---- prefetch next K-chunk into cache (global_prefetch_b8) ----
    if (k0 + KC < K) {
      __builtin_prefetch(Ablk + (size_t)(tid >> 1) * K + k0 + KC + (tid & 1) * 16, 0, 1);
      __builtin_prefetch(W + (size_t)(k0 + KC + (tid >> 3)) * N + tileN + (tid & 7) * 8, 0, 1);
    }
    __syncthreads();

    // ---- compute: 8 k-steps, fragments pipelined one step ahead ----
    v2f a0 = *(const v2f*)(&As[aBase]);
    v2f b00 = *(const v2f*)(&Bs[bBase + 0]);
    v2f b01 = *(const v2f*)(&Bs[bBase + 32]);
    v2f b02 = *(const v2f*)(&Bs[bBase + 64]);
    v2f b03 = *(const v2f*)(&Bs[bBase + 96]);
#pragma unroll
    for (int ks = 0; ks < KC / 4; ++ks) {
      v2f a1 = a0, b10 = b00, b11 = b01, b12 = b02, b13 = b03;
      if (ks + 1 < KC / 4) {
        a1 = *(const v2f*)(&As[aBase + (ks + 1) * 4]);
        const int bo = bBase + (ks + 1) * 2 * BPLD;
        b10 = *(const v2f*)(&Bs[bo + 0]);
        b11 = *(const v2f*)(&Bs[bo + 32]);
        b12 = *(const v2f*)(&Bs[bo + 64]);
        b13 = *(const v2f*)(&Bs[bo + 96]);
      }
      acc[0] = wmma4(a0, b00, acc[0]);
      acc[1] = wmma4(a0, b01, acc[1]);
      acc[2] = wmma4(a0, b02, acc[2]);
      acc[3] = wmma4(a0, b03, acc[3]);
      a0 = a1; b00 = b10; b01 = b11; b02 = b12; b03 = b13;
    }
    __syncthreads();
  }

  // Epilogue. C/D layout: VGPR i -> rows i (lanes 0-15) and i+8 (lanes 16-31).
#pragma unroll
  for (int i = 0; i < 8; ++i) {
    int m = tileM + rw + i + half * 8;
    size_t rowoff = (size_t)m * N;
#pragma unroll
    for (int nt = 0; nt < 4; ++nt) {
      int n = tileN + nt * 16 + lane_lo;
      float v = (acc[nt][i] + bias[n]) * preScale;
      if (act == 1) v = swish_f(v);
      if (res) v = res[rowoff + n] + resScale * v;
      out[rowoff + n] = v;
    }
  }
}

// ---------------------------------------------------------------------------
// Banded longformer attention via WMMA.
// One wave per 16-row t-tile: scores 16x80 = Q(16x64) @ K^T(64x80),
// masked softmax in LDS (writes probs), ctx = P(16x80) @ V(80x64),
// fused residual h += ctx. grid = (T/64, H, B), block = 128 (4 waves).
// q must already carry the 1/sqrt(DH) scale (applied in QKV epilogue).
// ---------------------------------------------------------------------------
__global__ __launch_bounds__(128) void attn_kernel(
    const float* __restrict__ Q, const float* __restrict__ Kg,
    const float* __restrict__ Vg, float* __restrict__ probs,
    float* __restrict__ h) {
  constexpr int SROW = 82;  // even (8B-aligned b64 P reads), banks distinct
  __shared__ float S[4][16 * SROW];

  const int wave = threadIdx.x >> 5;
  const int lane = threadIdx.x & 31;
  const int lane_lo = lane & 15;
  const int half = lane >> 4;
  const int t0 = blockIdx.x * 64 + wave * 16;
  const int hh = blockIdx.y;
  const int b = blockIdx.z;
  const int h0 = hh * DHh;
  float* Sw = &S[wave][0];

  // ---- scores: 5 accumulators of 16x16 over K = DH = 64 (16 k-steps) ----
  v8f sc[5] = {};
  const float* qbase = Q + ((size_t)b * Tt + (t0 + lane_lo)) * Dd + h0;
#pragma unroll 4
  for (int ks = 0; ks < 16; ++ks) {
    int kb = ks * 4 + half * 2;
    v2f a = *(const v2f*)(qbase + kb);
#pragma unroll
    for (int jt = 0; jt < 5; ++jt) {
      int tk = t0 - W1c + jt * 16 + lane_lo;
      v2f bb;
      if (tk >= 0 && tk < Tt) {
        bb = *(const v2f*)(Kg + ((size_t)b * Tt + tk) * Dd + h0 + kb);
      } else {
        bb.x = 0.f;
        bb.y = 0.f;
      }
      sc[jt] = wmma4(a, bb, sc[jt]);
    }
  }

  // spill scores to LDS
#pragma unroll
  for (int jt = 0; jt < 5; ++jt)
#pragma unroll
    for (int i = 0; i < 8; ++i)
      Sw[(i + half * 8) * SROW + jt * 16 + lane_lo] = sc[jt][i];
  __syncthreads();

  // ---- masked softmax, one lane per row; writes probs + P into LDS ----
  if (lane < 16) {
    int m = lane;
    int t = t0 + m;
    float* row = Sw + m * SROW;
    float mx = -3.0e38f;
    for (int w = 0; w < WBc; ++w) {
      int tk = t + w - W1c;
      float s = (tk >= 0 && tk < Tt) ? row[m + w] : -1.0e9f;
      mx = fmaxf(mx, s);
    }
    float sum = 0.f;
    for (int w = 0; w < WBc; ++w) {
      int tk = t + w - W1c;
      float s = (tk >= 0 && tk < Tt) ? row[m + w] : -1.0e9f;
      sum += expf(s - mx);
    }
    float inv = 1.f / sum;
    float* prow = probs + (((size_t)b * Tt + t) * Hh + hh) * WBc;
    for (int j = 0; j < 80; ++j) {
      int w = j - m;
      float p = 0.f;
      if (w >= 0 && w < WBc) {
        int tk = t + w - W1c;
        float s = (tk >= 0 && tk < Tt) ? row[j] : -1.0e9f;
        p = expf(s - mx) * inv;
        prow[w] = p;
      }
      row[j] = p;  // read of row[j] happened above in this iteration
    }
  }
  __syncthreads();

  // ---- ctx: 4 accumulators of 16x16 over K = 80 (20 k-steps) ----
  v8f ac[4] = {};
#pragma unroll 4
  for (int ks = 0; ks < 20; ++ks) {
    int kb = ks * 4 + half * 2;
    v2f a = *(const v2f*)(&Sw[lane_lo * SROW + kb]);
    int tk0 = t0 - W1c + kb;
    int tk1 = tk0 + 1;
#pragma unroll
    for (int nt = 0; nt < 4; ++nt) {
      int n = nt * 16 + lane_lo;
      v2f bb;
      bb.x = (tk0 >= 0 && tk0 < Tt)
                 ? Vg[((size_t)b * Tt + tk0) * Dd + h0 + n]
                 : 0.f;
      bb.y = (tk1 >= 0 && tk1 < Tt)
                 ? Vg[((size_t)b * Tt + tk1) * Dd + h0 + n]
                 : 0.f;
      ac[nt] = wmma4(a, bb, ac[nt]);
    }
  }

  // fused residual: h += ctx (each (b,t,d) owned by exactly one lane)
#pragma unroll
  for (int nt = 0; nt < 4; ++nt)
#pragma unroll
    for (int i = 0; i < 8; ++i) {
      int t = t0 + i + half * 8;
      int d = h0 + nt * 16 + lane_lo;
      size_t idx = ((size_t)b * Tt + t) * Dd + d;
      h[idx] += ac[nt][i];
    }
}

// ---------------------------------------------------------------------------
// GLU: out[bt,d] = in[bt,d] * sigmoid(in[bt,D+d]),  in stride CE=1024.
// Vectorized x4.
// ---------------------------------------------------------------------------
__global__ __launch_bounds__(256) void glu_kernel(const float* __restrict__ in,
                                                  float* __restrict__ out) {
  int idx = blockIdx.x * blockDim.x + threadIdx.x;  // over B*T*D/4
  int d4 = idx & 127;
  int bt = idx >> 7;
  float4 a = *(const float4*)(in + (size_t)bt * CEc + d4 * 4);
  float4 g = *(const float4*)(in + (size_t)bt * CEc + Dd + d4 * 4);
  float4 o;
  o.x = a.x / (1.0f + expf(-g.x));
  o.y = a.y / (1.0f + expf(-g.y));
  o.z = a.z / (1.0f + expf(-g.z));
  o.w = a.w / (1.0f + expf(-g.w));
  *(float4*)(out + (size_t)bt * Dd + d4 * 4) = o;
}

// ---------------------------------------------------------------------------
// Depthwise conv (KS=31, same-pad) + BN-scale + swish, fused.
// Each thread computes 4 consecutive t for one channel: 34-wide register
// window -> 34 loads instead of 124 per 4 outputs.
// ---------------------------------------------------------------------------
__global__ __launch_bounds__(256) void dwconv_kernel(
    const float* __restrict__ in, const float* __restrict__ dww,
    const float* __restrict__ dwb, const float* __restrict__ bng,
    const float* __restrict__ bnb, float* __restrict__ out) {
  int idx = blockIdx.x * blockDim.x + threadIdx.x;  // over B*(T/4)*D
  int d = idx & (Dd - 1);
  int t4 = (idx >> 9) & (Tt / 4 - 1);
  int b = idx >> 17;
  int t0 = t4 * 4;
  constexpr int PAD = (KSc - 1) / 2;  // 15
  float wbuf[KSc + 3];
#pragma unroll
  for (int j = 0; j < KSc + 3; ++j) {
    int tt = t0 - PAD + j;
    wbuf[j] = (tt >= 0 && tt < Tt) ? in[((size_t)b * Tt + tt) * Dd + d] : 0.f;
  }
  float o[4] = {0.f, 0.f, 0.f, 0.f};
#pragma unroll
  for (int k = 0; k < KSc; ++k) {
    float wk = dww[k * Dd + d];
#pragma unroll
    for (int i = 0; i < 4; ++i) o[i] += wbuf[i + k] * wk;
  }
  const float scale = rsqrtf(1.0f + 1e-5f);
  float db = dwb[d], bg = bng[d], bbeta = bnb[d];
#pragma unroll
  for (int i = 0; i < 4; ++i) {
    float v = (o[i] + db) * scale * bg + bbeta;
    out[((size_t)b * Tt + t0 + i) * Dd + d] = swish_f(v);
  }
}

// ---------------------------------------------------------------------------
// Host-side orchestration.
// ---------------------------------------------------------------------------
extern "C" void kernel_launch(void* const* d_in, const int* in_sizes, int n_in,
                              void* d_out, int out_size, void* d_ws,
                              size_t ws_size, hipStream_t stream) {
  (void)in_sizes; (void)n_in; (void)out_size; (void)ws_size;

  const float* x = (const float*)d_in[0];
  const float* ff1_ln_g = (const float*)d_in[1];
  const float* ff1_ln_b = (const float*)d_in[2];
  const float* ff1_w1 = (const float*)d_in[3];
  const float* ff1_b1 = (const float*)d_in[4];
  const float* ff1_w2 = (const float*)d_in[5];
  const float* ff1_b2 = (const float*)d_in[6];
  const float* ff2_ln_g = (const float*)d_in[7];
  const float* ff2_ln_b = (const float*)d_in[8];
  const float* ff2_w1 = (const float*)d_in[9];
  const float* ff2_b1 = (const float*)d_in[10];
  const float* ff2_w2 = (const float*)d_in[11];
  const float* ff2_b2 = (const float*)d_in[12];
  const float* wq = (const float*)d_in[13];
  const float* bq = (const float*)d_in[14];
  const float* wk = (const float*)d_in[15];
  const float* bk = (const float*)d_in[16];
  const float* wv = (const float*)d_in[17];
  const float* bv = (const float*)d_in[18];
  const float* conv_ln_g = (const float*)d_in[19];
  const float* conv_ln_b = (const float*)d_in[20];
  const float* conv_pw1_w = (const float*)d_in[21];
  const float* conv_pw1_b = (const float*)d_in[22];
  const float* conv_dw_w = (const float*)d_in[23];
  const float* conv_dw_b = (const float*)d_in[24];
  const float* conv_bn_g = (const float*)d_in[25];
  const float* conv_bn_b = (const float*)d_in[26];
  const float* conv_pw2_w = (const float*)d_in[27];
  const float* conv_pw2_b = (const float*)d_in[28];
  const float* fin_ln_g = (const float*)d_in[29];
  const float* fin_ln_b = (const float*)d_in[30];

  const size_t BTD = (size_t)Bb * Tt * Dd;      // 2,097,152
  const size_t BTFF = (size_t)Bb * Tt * FFc;    // 8,388,608

  float* h = (float*)d_out;            // running residual == final out region
  float* probs = h + BTD;              // second output region

  float* ws = (float*)d_ws;
  float* buf_ln = ws;                  // BTD
  float* buf_ff = buf_ln + BTD;        // BTFF (also holds CE-wide pw1 out)
  float* q = buf_ff + BTFF;            // BTD
  float* k = q + BTD;                  // BTD
  float* v = k + BTD;                  // BTD

  const int M = Bb * Tt;  // 4096
  dim3 blk256(256), blk128(128);
  dim3 gLN(M / 8);                       // 8 rows per 256-thread block
  dim3 gElem4((unsigned)(BTD / 4 / 256));
  const float qscale = 0.125f;           // 1/sqrt(DH)

  auto gemm = [&](const float* A, const float* W, const float* bias,
                  float* out, int N, int K, int act, float preScale,
                  const float* res, float resScale) {
    dim3 grid(N / 64, M / 128);
    gemm_wmma_kernel<<<grid, blk256, 0, stream>>>(A, W, bias, out, M, N, K,
                                                  act, preScale, res, resScale);
  };

  // ---- FF1 (half-step): h = x + 0.5 * ff(x) ----
  ln_kernel<<<gLN, blk256, 0, stream>>>(x, ff1_ln_g, ff1_ln_b, buf_ln, M);
  gemm(buf_ln, ff1_w1, ff1_b1, buf_ff, FFc, Dd, /*act=*/1, 1.f, nullptr, 0.f);
  gemm(buf_ff, ff1_w2, ff1_b2, h, Dd, FFc, /*act=*/0, 1.f, x, 0.5f);

  // ---- Longformer attention: h += ctx ----
  gemm(h, wq, bq, q, Dd, Dd, 0, qscale, nullptr, 0.f);
  gemm(h, wk, bk, k, Dd, Dd, 0, 1.f, nullptr, 0.f);
  gemm(h, wv, bv, v, Dd, Dd, 0, 1.f, nullptr, 0.f);
  {
    dim3 grid(Tt / 64, Hh, Bb);
    attn_kernel<<<grid, blk128, 0, stream>>>(q, k, v, probs, h);
  }

  // ---- Conv module: h += conv(h)  (q,k reused as scratch) ----
  ln_kernel<<<gLN, blk256, 0, stream>>>(h, conv_ln_g, conv_ln_b, buf_ln, M);
  gemm(buf_ln, conv_pw1_w, conv_pw1_b, buf_ff, CEc, Dd, 0, 1.f, nullptr, 0.f);
  glu_kernel<<<gElem4, blk256, 0, stream>>>(buf_ff, q);
  dwconv_kernel<<<gElem4, blk256, 0, stream>>>(q, conv_dw_w, conv_dw_b,
                                               conv_bn_g, conv_bn_b, k);
  gemm(k, conv_pw2_w, conv_pw2_b, h, Dd, Dd, 0, 1.f, h, 1.f);

  // ---- FF2 (half-step): h += 0.5 * ff(h) ----
  ln_kernel<<<gLN, blk256, 0, stream>>>(h, ff2_ln_g, ff2_ln_b, buf_ln, M);
  gemm(buf_ln, ff2_w1, ff2_b1, buf_ff, FFc, Dd, 1, 1.f, nullptr, 0.f);
  gemm(buf_ff, ff2_w2, ff2_b2, h, Dd, FFc, 0, 1.f, h, 0.5f);

  // ---- Final LN, in place over the out region ----
  ln_kernel<<<gLN, blk256, 0, stream>>>(h, fin_ln_g, fin_ln_b, h, M);
}